// BERT_CRF_11269994185172
// MI455X (gfx1250) — compile-verified
//
#include <hip/hip_runtime.h>
#include <hip/hip_bf16.h>

typedef float v2f __attribute__((ext_vector_type(2)));
typedef float v8f __attribute__((ext_vector_type(8)));

#define Bn 64
#define Tn 512
#define Hn 768
#define Ln 9

// ---------------------------------------------------------------------------
// Kernel 1: emissions = word2vec @ W^T + b, padded to 16 label columns.
// One wave computes one 16x16 output tile via V_WMMA_F32_16X16X4_F32,
// K-loop over H=768 in steps of 4 (192 WMMAs / tile).
//
// A fragment (16x4 f32):  lane<16: {A[m][k],A[m][k+1]}, lane>=16: {A[m][k+2],A[m][k+3]},
//   m = lane&15  -> contiguous float2 at X + m*H + k + 2*(lane>>4)
// B fragment (4x16 f32):  lane n=lane&15 holds column n:
//   {B[k'][n],B[k'+1][n]} = {W[n][k'],W[n][k'+1]}  -> contiguous float2 in W.
//   Columns 9..15 are never consumed downstream, so we clamp the W row index
//   (branch-free) and let the pad columns hold garbage.
// C/D (16x16 f32): VGPR v, lane: row = v + 8*(lane>>4), col = lane&15
// ---------------------------------------------------------------------------
__global__ __launch_bounds__(128)
void emissions_wmma_kernel(const float* __restrict__ X,     // (32768, 768)
                           const float* __restrict__ W,     // (9, 768)
                           const float* __restrict__ bias,  // (9)
                           float* __restrict__ emPad)       // (32768, 16)
{
    const int wave = blockIdx.x * (blockDim.x >> 5) + (threadIdx.x >> 5);
    const int lane = threadIdx.x & 31;
    const int n    = lane & 15;                 // column / row-in-tile index
    const int kh   = lane >> 4;                 // K half select (0 or 1)
    const int nc   = (n < Ln) ? n : (Ln - 1);   // clamped label row (pad cols = don't-care)

    const long r0 = (long)wave * 16;
    const float* aPtr = X + (r0 + n) * Hn + 2 * kh;   // A stream (per-lane, unique 8B)
    const float* bPtr = W + nc * Hn + 2 * kh;         // W row (L0/L2 resident)

    v8f c = {0.f, 0.f, 0.f, 0.f, 0.f, 0.f, 0.f, 0.f};

    // K in blocks of 64; prefetch the next A block into the near caches.
    for (int k0 = 0; k0 < Hn; k0 += 64) {
        if (k0 + 64 < Hn) {
            __builtin_prefetch(aPtr + k0 + 64, 0, 3);   // -> global_prefetch_b8
        }
#pragma unroll
        for (int kk = 0; kk < 64; kk += 4) {
            const int k = k0 + kk;
            v2f a = __builtin_nontemporal_load((const v2f*)(aPtr + k));
            v2f b = *(const v2f*)(bPtr + k);
            // D = A x B + C   (v_wmma_f32_16x16x4_f32)
            c = __builtin_amdgcn_wmma_f32_16x16x4_f32(
                    /*neg_a=*/false, a, /*neg_b=*/false, b,
                    /*c_mod=*/(short)0, c, /*reuse_a=*/false, /*reuse_b=*/false);
        }
    }

    const float bv = bias[nc];        // pad columns hold garbage; never read
    float* outBase = emPad + r0 * 16 + n;
#pragma unroll
    for (int v = 0; v < 8; ++v) {
        outBase[(v + 8 * kh) * 16] = c[v] + bv;
    }
}

// ---------------------------------------------------------------------------
// Kernel 2: CRF NLL per batch. One wave (32 lanes) per batch element.
// Phase 1: numerator (parallel over t, branch-free mask-multiply, xor-tree reduce).
// Phase 2: forward algorithm; lane i holds alpha[i] (i<9), shfl-gather per step.
// ---------------------------------------------------------------------------
__global__ __launch_bounds__(32)
void crf_nll_kernel(const float* __restrict__ emPad,   // (B, T, 16)
                    const int*   __restrict__ mask,    // (B, T)
                    const int*   __restrict__ label,   // (B, T)
                    const float* __restrict__ startT,  // (9)
                    const float* __restrict__ endT,    // (9)
                    const float* __restrict__ trans,   // (9, 9)
                    float* __restrict__ lossPerBatch)  // (B)
{
    const int b    = blockIdx.x;
    const int lane = threadIdx.x;
    const float* em = emPad + (long)b * Tn * 16;
    const int*   mk = mask  + (long)b * Tn;
    const int*   tg = label + (long)b * Tn;

    // ---- numerator: start + em[0,tag0] + sum_t mask*(trans + em) + end[last]
    float s = 0.f;
    for (int t = 1 + lane; t < Tn; t += 32) {
        const int pt = tg[t - 1], ct = tg[t];        // always in [0,9): loads valid
        const float mf = (mk[t] != 0) ? 1.f : 0.f;   // branch-free gate
        s += mf * (trans[pt * Ln + ct] + em[t * 16 + ct]);
    }
    int cnt = 0;
    for (int t = lane; t < Tn; t += 32) cnt += (mk[t] != 0);
#pragma unroll
    for (int off = 16; off >= 1; off >>= 1) {
        s   += __shfl_xor(s, off, 32);
        cnt += __shfl_xor(cnt, off, 32);
    }
    const int t0tag = tg[0];
    float num = startT[t0tag] + em[t0tag] + s;
    num += endT[tg[cnt - 1]];

    // ---- denominator: forward algorithm in log space
    const float NEG = -3.0e38f;
    const int i  = lane;
    const int ic = (i < Ln) ? i : (Ln - 1);   // clamped state (lanes >=9 do dummy work)
    const int ie = i & 15;                    // clamped emission column

    float tci[Ln];                            // trans[j][i] for this lane's state
#pragma unroll
    for (int j = 0; j < Ln; ++j) tci[j] = trans[j * Ln + ic];

    float alpha = (i < Ln) ? (startT[ic] + em[ic]) : NEG;

    for (int t = 1; t < Tn; ++t) {
        const float emi = em[t * 16 + ie];    // garbage for lanes >=9 (discarded)
        float xj[Ln];
        float m = NEG;
#pragma unroll
        for (int j = 0; j < Ln; ++j) {
            const float aj = __shfl(alpha, j, 32);
            xj[j] = aj + tci[j];
            m = fmaxf(m, xj[j]);
        }
        float ssum = 0.f;
#pragma unroll
        for (int j = 0; j < Ln; ++j) ssum += __expf(xj[j] - m);
        float nxt = m + __logf(ssum) + emi;
        if (i >= Ln) nxt = NEG;               // v_cndmask
        if (mk[t])   alpha = nxt;             // v_cndmask
    }

    float v = (i < Ln) ? (alpha + endT[ic]) : NEG;
    float mm = v;
#pragma unroll
    for (int off = 16; off >= 1; off >>= 1) mm = fmaxf(mm, __shfl_xor(mm, off, 32));
    float se = (i < Ln) ? __expf(v - mm) : 0.f;
#pragma unroll
    for (int off = 16; off >= 1; off >>= 1) se += __shfl_xor(se, off, 32);
    const float den = mm + __logf(se);

    if (lane == 0) lossPerBatch[b] = den - num;   // -llh contribution
}

// ---------------------------------------------------------------------------
// Kernel 3: mean over batches -> scalar loss
// ---------------------------------------------------------------------------
__global__ __launch_bounds__(32)
void finalize_kernel(const float* __restrict__ lossPerBatch, float* __restrict__ out)
{
    const int lane = threadIdx.x;
    float s = 0.f;
    for (int b = lane; b < Bn; b += 32) s += lossPerBatch[b];
#pragma unroll
    for (int off = 16; off >= 1; off >>= 1) s += __shfl_xor(s, off, 32);
    if (lane == 0) out[0] = s / (float)Bn;
}

extern "C" void kernel_launch(void* const* d_in, const int* in_sizes, int n_in,
                              void* d_out, int out_size, void* d_ws, size_t ws_size,
                              hipStream_t stream) {
    // setup_inputs order:
    // 0 length(int,64) 1 word2vec(f32,B*T*H) 2 mask(int,B*T) 3 label(int,B*T)
    // 4 W(f32,9*768) 5 b(f32,9) 6 start_transitions(9) 7 end_transitions(9) 8 transitions(81)
    const float* word2vec = (const float*)d_in[1];
    const int*   mask     = (const int*)  d_in[2];
    const int*   label    = (const int*)  d_in[3];
    const float* W        = (const float*)d_in[4];
    const float* bias     = (const float*)d_in[5];
    const float* startT   = (const float*)d_in[6];
    const float* endT     = (const float*)d_in[7];
    const float* trans    = (const float*)d_in[8];

    float* emPad = (float*)d_ws;                     // B*T*16 floats = 2 MB
    float* lossB = emPad + (size_t)Bn * Tn * 16;     // 64 floats

    const int rows  = Bn * Tn;        // 32768
    const int tiles = rows / 16;      // 2048 waves
    const int wavesPerBlock = 4;      // 128 threads
    emissions_wmma_kernel<<<tiles / wavesPerBlock, wavesPerBlock * 32, 0, stream>>>(
        word2vec, W, bias, emPad);

    crf_nll_kernel<<<Bn, 32, 0, stream>>>(emPad, mask, label, startT, endT, trans, lossB);

    finalize_kernel<<<1, 32, 0, stream>>>(lossB, (float*)d_out);
}